// ConceptFlow_55843164782666
// MI455X (gfx1250) — compile-verified
//
#include <hip/hip_runtime.h>
#include <hip/hip_bf16.h>

#define D 100

typedef __attribute__((ext_vector_type(2))) float v2f;
typedef __attribute__((ext_vector_type(8))) float v8f;

// Order-preserving float <-> uint bijection so segment-max can use native
// global_atomic_max_u32 (single atomic, no CAS loop).
__device__ __forceinline__ unsigned enc_f32(float f) {
    unsigned b = __float_as_uint(f);
    return (b & 0x80000000u) ? ~b : (b | 0x80000000u);
}
__device__ __forceinline__ float dec_f32(unsigned u) {
    unsigned b = (u & 0x80000000u) ? (u & 0x7FFFFFFFu) : ~u;
    return __uint_as_float(b);
}

// ---------------------------------------------------------------------------
// K0: v2[i] = sum_k W_rel[i,k]*w2[k]; v3[i] = sum_k W_rel[100+i,k]*w2[k];
//     c = b_rel . w2      (w2 = W_attn[100:200])
// ---------------------------------------------------------------------------
__global__ void prep_kernel(const float* __restrict__ Wrel,
                            const float* __restrict__ brel,
                            const float* __restrict__ Wattn,
                            float* __restrict__ ws) {
    int i = threadIdx.x;
    if (i < D) {
        float a2 = 0.f, a3 = 0.f;
        for (int k = 0; k < D; ++k) {
            float w2 = Wattn[D + k];
            a2 += Wrel[i * D + k] * w2;
            a3 += Wrel[(D + i) * D + k] * w2;
        }
        ws[i]     = a2;   // v2
        ws[D + i] = a3;   // v3
    }
    if (i == D) {
        float c = 0.f;
        for (int k = 0; k < D; ++k) c += brel[k] * Wattn[D + k];
        ws[2 * D] = c;
    }
}

// ---------------------------------------------------------------------------
// K1: zero h, denom; m_u32 = 0 (encodes below any real encoded float)
// ---------------------------------------------------------------------------
__global__ void init_kernel(float* __restrict__ h, float* __restrict__ denom,
                            unsigned* __restrict__ m_u32, int n, long total) {
    long idx = (long)blockIdx.x * blockDim.x + threadIdx.x;
    if (idx < total) h[idx] = 0.f;
    if (idx < n) { denom[idx] = 0.f; m_u32[idx] = 0u; }
}

// ---------------------------------------------------------------------------
// K2: z = x @ W_fc via V_WMMA_F32_16X16X4_F32.
// Block = 256 threads (8 waves); block owns a 16-row tile of x (staged in
// LDS); waves 0..6 each produce one 16-col tile (cols 96..111 masked at
// store).  For the partial tile the B-load column is CLAMPED instead of
// guarded: B column n only feeds C column n, and those lanes never store,
// so the inner loop has no exec-mask branching at all.
// A layout (16x4 f32): lanes 0-15 row M=lane, v0=K0,v1=K1; lanes 16-31 K2,K3.
// B layout (4x16 f32): v0 = rows K0(l0-15)/K2(l16-31), v1 = K1/K3, col=lane&15.
// C layout: vgpr r -> M = r (+8 for upper lanes), N = lane&15.
// ---------------------------------------------------------------------------
__global__ __launch_bounds__(256) void gemm_kernel(const float* __restrict__ x,
                                                   const float* __restrict__ Wfc,
                                                   float* __restrict__ z, int n) {
    __shared__ float xs[16 * D];
    const int mt  = blockIdx.x;
    const int tid = threadIdx.x;
    const long base = (long)mt * 16 * D;
    const long lim  = (long)n * D;
    for (int i = tid; i < 16 * D; i += 256)
        xs[i] = (base + i < lim) ? x[base + i] : 0.f;
    __syncthreads();

    const int wave = tid >> 5;
    const int lane = tid & 31;
    if (wave >= 7) return;

    const int n0   = wave * 16;
    const int col  = n0 + (lane & 15);
    const int colc = (col < D) ? col : (D - 1);   // clamped load column
    const int kk   = (lane < 16) ? 0 : 2;
    const int arow = lane & 15;

    v8f c = {};
    for (int k = 0; k < D; k += 4) {
        v2f a, b;
        a.x = xs[arow * D + k + kk];
        a.y = xs[arow * D + k + kk + 1];
        b.x = Wfc[(k + kk) * D + colc];
        b.y = Wfc[(k + kk + 1) * D + colc];
        c = __builtin_amdgcn_wmma_f32_16x16x4_f32(false, a, false, b,
                                                  (short)0, c, false, false);
    }
    if (col < D) {
        const int rbase = mt * 16 + ((lane < 16) ? 0 : 8);
        for (int r = 0; r < 8; ++r) {
            int row = rbase + r;
            if (row < n) z[(long)row * D + col] = c[r];
        }
    }
}

// ---------------------------------------------------------------------------
// K3: per-node scalars s1[n] = z[n].w1, s3[n] = z[n].v3  (wave per node)
// ---------------------------------------------------------------------------
__global__ __launch_bounds__(256) void node_kernel(const float* __restrict__ z,
                                                   const float* __restrict__ Wattn,
                                                   const float* __restrict__ v3,
                                                   float* __restrict__ s1,
                                                   float* __restrict__ s3, int n) {
    __shared__ float w1s[D], v3s[D];
    int tid = threadIdx.x;
    if (tid < D) { w1s[tid] = Wattn[tid]; v3s[tid] = v3[tid]; }
    __syncthreads();
    int wave = tid >> 5, lane = tid & 31;
    int node = blockIdx.x * 8 + wave;
    if (node >= n) return;
    const float* zp = z + (long)node * D;
    float a1 = 0.f, a3 = 0.f;
    for (int c = lane; c < D; c += 32) {
        float v = zp[c];
        a1 += v * w1s[c];
        a3 += v * v3s[c];
    }
    for (int off = 16; off; off >>= 1) {
        a1 += __shfl_xor(a1, off, 32);
        a3 += __shfl_xor(a3, off, 32);
    }
    if (lane == 0) { s1[node] = a1; s3[node] = a3; }
}

// ---------------------------------------------------------------------------
// K4: per-edge logit  e = leaky_relu(s1[src] + edge_h.v2 + s3[dst] + c)
//     + segment max via encoded atomicMax           (wave per edge)
// ---------------------------------------------------------------------------
__global__ __launch_bounds__(256) void edge_score_kernel(
        const float* __restrict__ edge_h, const int* __restrict__ src,
        const int* __restrict__ dst, const float* __restrict__ v2,
        const float* __restrict__ cscal, const float* __restrict__ s1,
        const float* __restrict__ s3, float* __restrict__ escore,
        unsigned* __restrict__ m_u32, int ne) {
    __shared__ float v2s[D];
    int tid = threadIdx.x;
    if (tid < D) v2s[tid] = v2[tid];
    __syncthreads();
    int wave = tid >> 5, lane = tid & 31;
    int e = blockIdx.x * 8 + wave;
    if (e >= ne) return;
    const float* ep = edge_h + (long)e * D;
    float t = 0.f;
    for (int c = lane; c < D; c += 32) t += ep[c] * v2s[c];
    for (int off = 16; off; off >>= 1) t += __shfl_xor(t, off, 32);
    if (lane == 0) {
        int dn = dst[e];
        float s  = s1[src[e]] + t + s3[dn] + cscal[0];
        float ls = (s > 0.f) ? s : 0.01f * s;
        escore[e] = ls;
        atomicMax(&m_u32[dn], enc_f32(ls));
    }
}

// ---------------------------------------------------------------------------
// K5: ex = exp(e - m[dst]); denom[dst] += ex        (thread per edge)
// ---------------------------------------------------------------------------
__global__ void edge_exp_kernel(const int* __restrict__ dst,
                                const unsigned* __restrict__ m_u32,
                                float* __restrict__ escore,
                                float* __restrict__ denom, int ne) {
    int e = blockIdx.x * blockDim.x + threadIdx.x;
    if (e >= ne) return;
    int dn = dst[e];
    float m  = dec_f32(m_u32[dn]);
    float ex = __expf(escore[e] - m);
    escore[e] = ex;
    atomicAdd(&denom[dn], ex);
}

// ---------------------------------------------------------------------------
// K6: h[dst] += (ex/denom[dst]) * z[src]            (wave per edge)
// z and h are both ~20MB -> L2-resident on the 192MB L2.
// ---------------------------------------------------------------------------
__global__ __launch_bounds__(256) void scatter_kernel(
        const float* __restrict__ z, const int* __restrict__ src,
        const int* __restrict__ dst, const float* __restrict__ ex,
        const float* __restrict__ denom, float* __restrict__ h, int ne) {
    int tid = threadIdx.x;
    int wave = tid >> 5, lane = tid & 31;
    int e = blockIdx.x * 8 + wave;
    if (e >= ne) return;
    int s = src[e], dn = dst[e];
    float alpha = ex[e] / denom[dn];
    const float* zp = z + (long)s * D;
    float* hp = h + (long)dn * D;
    for (int c = lane; c < D; c += 32)
        atomicAdd(&hp[c], alpha * zp[c]);
}

// ---------------------------------------------------------------------------
extern "C" void kernel_launch(void* const* d_in, const int* in_sizes, int n_in,
                              void* d_out, int out_size, void* d_ws, size_t ws_size,
                              hipStream_t stream) {
    const float* x      = (const float*)d_in[0];
    const float* edge_h = (const float*)d_in[1];
    const int*   src    = (const int*)d_in[2];
    const int*   dst    = (const int*)d_in[3];
    const float* Wfc    = (const float*)d_in[4];
    const float* Wrel   = (const float*)d_in[5];
    const float* brel   = (const float*)d_in[6];
    const float* Wattn  = (const float*)d_in[7];
    float* h = (float*)d_out;

    const int n  = in_sizes[0] / D;   // 50000 nodes
    const int ne = in_sizes[2];       // 400000 edges

    // workspace layout (floats)
    float* ws     = (float*)d_ws;
    float* v2     = ws;               // [D]
    float* v3     = ws + D;           // [D]
    float* cs     = ws + 2 * D;       // [1]
    float* s1     = ws + 256;         // [n]
    float* s3     = s1 + n;           // [n]
    float* denom  = s3 + n;           // [n]
    unsigned* m_u = (unsigned*)(denom + n);   // [n]
    float* escore = (float*)(m_u + n);        // [ne]
    float* z      = escore + ne;              // [n*D]

    prep_kernel<<<1, 128, 0, stream>>>(Wrel, brel, Wattn, ws);

    const long total = (long)n * D;
    init_kernel<<<(int)((total + 255) / 256), 256, 0, stream>>>(h, denom, m_u, n, total);

    const int mtiles = (n + 15) / 16;
    gemm_kernel<<<mtiles, 256, 0, stream>>>(x, Wfc, z, n);

    node_kernel<<<(n + 7) / 8, 256, 0, stream>>>(z, Wattn, v3, s1, s3, n);

    edge_score_kernel<<<(ne + 7) / 8, 256, 0, stream>>>(edge_h, src, dst, v2, cs,
                                                        s1, s3, escore, m_u, ne);

    edge_exp_kernel<<<(ne + 255) / 256, 256, 0, stream>>>(dst, m_u, escore, denom, ne);

    scatter_kernel<<<(ne + 7) / 8, 256, 0, stream>>>(z, src, dst, escore, denom, h, ne);
}